// CrumbReconstructor_44281112821816
// MI455X (gfx1250) — compile-verified
//
#include <hip/hip_runtime.h>

typedef float v2f __attribute__((ext_vector_type(2)));
typedef float v4f __attribute__((ext_vector_type(4)));
typedef float v8f __attribute__((ext_vector_type(8)));

#define NBLK   802816          // total 8-float keys = 64*14*14*512/8
#define NPAIRS (NBLK / 32)     // 25088 pairs of 16-key tiles (32 keys / wave iter)
#define MROWS  256
#define LBLK   8

__launch_bounds__(256, 2)
__global__ void vq_wmma_f32_kernel(const float* __restrict__ x,
                                   const float* __restrict__ mem,
                                   float* __restrict__ out)
{
    // LDS staging: raw codebook (gather), B-operand swizzle, row norms
    __shared__ __align__(16) float ldsMem[MROWS * LBLK];   // 8 KB
    __shared__ v4f               ldsB[16 * 32];            // 8 KB
    __shared__ float             ldsNorm[MROWS];           // 1 KB

    const int tid = threadIdx.x;

    // (a) raw codebook copy (for the final gather)
    for (int i = tid; i < MROWS * LBLK; i += 256) ldsMem[i] = mem[i];

    // (b) B-operand layout: entry i = (t*32 + lane).
    //     B (4x16 f32, 2 VGPRs): v0 lane<16 -> B[K=0][N=lane] = mem[t*16+lane][c0+0]
    //                            v1 lane<16 -> K=1 -> col c0+1 ; lanes>=16 -> K=2,3 (c0=2)
    //     Pack both K-halves (cols {c0,c0+1} and {c0+4,c0+5}) into one float4.
    for (int i = tid; i < 16 * 32; i += 256) {
        int lane = i & 31;
        int row  = ((i >> 5) << 4) + (lane & 15);   // t*16 + lane%16
        int c0   = (lane >> 4) << 1;                // 0 (lanes 0-15) or 2 (lanes 16-31)
        const float* r = mem + row * LBLK + c0;
        v4f b;
        b.x = r[0]; b.y = r[1];      // K-step 0 operand pair
        b.z = r[4]; b.w = r[5];      // K-step 1 operand pair
        ldsB[i] = b;
    }

    // (c) ||m_j||^2
    for (int i = tid; i < MROWS; i += 256) {
        const float* r = mem + i * LBLK;
        float s = 0.f;
        #pragma unroll
        for (int j = 0; j < LBLK; ++j) s = fmaf(r[j], r[j], s);
        ldsNorm[i] = s;
    }
    __syncthreads();

    const int lane   = tid & 31;
    const int wave   = (blockIdx.x << 3) + (tid >> 5);
    const int nWaves = gridDim.x << 3;
    const int col    = lane & 15;
    const int half   = lane >> 4;

    for (int pair = wave; pair < NPAIRS; pair += nWaves) {
        // Two 16-key A tiles (32 keys = 256 contiguous floats).
        // A-operand (16x4 f32): lane l<16 holds key row l, K={0,1}; l>=16 holds K={2,3}.
        const float* aBase = x + (size_t)pair * 256 + col * LBLK + half * 2;
        v2f aLo0 = *(const v2f*)(aBase);         // tile 0, key elems {0,1}/{2,3}
        v2f aHi0 = *(const v2f*)(aBase + 4);     // tile 0, key elems {4,5}/{6,7}
        v2f aLo1 = *(const v2f*)(aBase + 128);   // tile 1
        v2f aHi1 = *(const v2f*)(aBase + 132);

        // speculative prefetch of the next pair this wave will touch
        __builtin_prefetch(x + ((size_t)pair + nWaves) * 256 + lane * 8, 0, 1);

        float best0[8], best1[8];
        int   bidx0[8], bidx1[8];
        #pragma unroll
        for (int r = 0; r < 8; ++r) {
            best0[r] = 3.4e38f; bidx0[r] = 0;
            best1[r] = 3.4e38f; bidx1[r] = 0;
        }

        #pragma unroll 4
        for (int t = 0; t < 16; ++t) {                 // 16 codebook column tiles
            v4f   b4  = ldsB[t * 32 + lane];           // shared by both A tiles
            float nrm = ldsNorm[t * 16 + col];
            v2f b0 = { b4.x, b4.y };
            v2f b1 = { b4.z, b4.w };

            // D = A[:, :4] x B[:4, :] ; then accumulate A[:, 4:8] x B[4:8, :]
            v8f c0 = {};
            c0 = __builtin_amdgcn_wmma_f32_16x16x4_f32(false, aLo0, false, b0,
                                                       (short)0, c0, false, false);
            c0 = __builtin_amdgcn_wmma_f32_16x16x4_f32(false, aHi0, false, b1,
                                                       (short)0, c0, false, false);
            v8f c1 = {};
            c1 = __builtin_amdgcn_wmma_f32_16x16x4_f32(false, aLo1, false, b0,
                                                       (short)0, c1, false, false);
            c1 = __builtin_amdgcn_wmma_f32_16x16x4_f32(false, aHi1, false, b1,
                                                       (short)0, c1, false, false);

            // D slot r, this lane -> key (half*8 + r), codebook row n
            int n = t * 16 + col;
            #pragma unroll
            for (int r = 0; r < 8; ++r) {
                float s0 = fmaf(-2.0f, c0[r], nrm);    // ||m||^2 - 2 k.m
                if (s0 < best0[r]) { best0[r] = s0; bidx0[r] = n; }
                float s1 = fmaf(-2.0f, c1[r], nrm);
                if (s1 < best1[r]) { best1[r] = s1; bidx1[r] = n; }
            }
        }

        // Butterfly min-reduce over the 16 lanes sharing each key slot.
        // (xor masks < 16 never cross the half boundary.)
        #pragma unroll
        for (int off = 8; off >= 1; off >>= 1) {
            #pragma unroll
            for (int r = 0; r < 8; ++r) {
                float ob0 = __shfl_xor(best0[r], off, 32);
                int   oi0 = __shfl_xor(bidx0[r], off, 32);
                if (ob0 < best0[r] || (ob0 == best0[r] && oi0 < bidx0[r])) {
                    best0[r] = ob0; bidx0[r] = oi0;
                }
                float ob1 = __shfl_xor(best1[r], off, 32);
                int   oi1 = __shfl_xor(bidx1[r], off, 32);
                if (ob1 < best1[r] || (ob1 == best1[r] && oi1 < bidx1[r])) {
                    best1[r] = ob1; bidx1[r] = oi1;
                }
            }
        }

        // Gather winner rows from LDS and store (2 x 512 B contiguous per wave).
        // lane -> key (half*8 + lane%8), elements [(lane/8 & 1)*4, +4)
        int slot = lane & 7;
        int e0   = ((lane >> 3) & 1) << 2;
        int key  = (half << 3) + slot;
        float* outBase = out + (size_t)pair * 256 + key * LBLK + e0;

        v4f v0 = *(const v4f*)(ldsMem + bidx0[slot] * LBLK + e0);
        *(v4f*)(outBase) = v0;
        v4f v1 = *(const v4f*)(ldsMem + bidx1[slot] * LBLK + e0);
        *(v4f*)(outBase + 128) = v1;
    }
}

extern "C" void kernel_launch(void* const* d_in, const int* in_sizes, int n_in,
                              void* d_out, int out_size, void* d_ws, size_t ws_size,
                              hipStream_t stream) {
    (void)in_sizes; (void)n_in; (void)d_ws; (void)ws_size; (void)out_size;
    const float* x   = (const float*)d_in[0];   // [64,14,14,512] f32
    const float* mem = (const float*)d_in[1];   // [256,8] f32
    float*       out = (float*)d_out;           // [64,14,14,512] f32

    dim3 grid(512), block(256);                 // 4096 waves grid-striding 25088 pairs
    hipLaunchKernelGGL(vq_wmma_f32_kernel, grid, block, 0, stream, x, mem, out);
}